// IntensityTransformation_44916767981885
// MI455X (gfx1250) — compile-verified
//
#include <hip/hip_runtime.h>
#include <cstdint>
#include <cstddef>

// out_k[b,c,h,w] = tf_k[b,c, round(255*img[b,c,h,w])]  for k=1..3
// Pure bandwidth-bound LUT gather: ~403MB of HBM traffic, ~17us floor @ 23.3TB/s.
// gfx1250 paths used: global_load_async_to_lds_b32 (+ s_wait_asynccnt) for LUT
// staging, ds_load_b32 gathers, nontemporal B128 streaming loads/stores.

typedef float v4f __attribute__((ext_vector_type(4)));
typedef __attribute__((address_space(3))) float lds_f;

static constexpr int kH = 1024, kW = 1024, kB = 8, kC = 3;
static constexpr int kHW = kH * kW;                    // 1,048,576 px per slice
static constexpr int kSlices = kB * kC;                // 24
static constexpr long long kTot = (long long)kSlices * kHW; // 25,165,824 (one output's elems)
static constexpr int kVecPerSlice = kHW / 4;           // 262,144 float4s per slice

__global__ __launch_bounds__(256)
void IntensityTransformation_lut3_kernel(const float* __restrict__ img,
                                         const float* __restrict__ tf1,
                                         const float* __restrict__ tf2,
                                         const float* __restrict__ tf3,
                                         float* __restrict__ out) {
    __shared__ float lut[3 * 256];   // 3KB: tf1 | tf2 | tf3 for this (b,c) slice

    const int slice = blockIdx.y;    // 0..23 = flat (b,c)
    const int t     = threadIdx.x;   // 0..255

    // ---- Stage the three 256-entry LUTs into LDS via gfx1250 async DMA ----
    // One b32 element per lane per table: vdst gives the LDS address, saddr + vaddr the global address.
    {
        const unsigned lbase = (unsigned)(unsigned long long)(lds_f*)lut; // LDS byte offset
        const unsigned goff  = (unsigned)(t * 4);                         // byte offset in table
        const float* s1 = tf1 + (size_t)slice * 256;
        const float* s2 = tf2 + (size_t)slice * 256;
        const float* s3 = tf3 + (size_t)slice * 256;
        const unsigned l1 = lbase + goff;
        const unsigned l2 = lbase + 1024u + goff;
        const unsigned l3 = lbase + 2048u + goff;
        asm volatile("global_load_async_to_lds_b32 %0, %1, %2"
                     :: "v"(l1), "v"(goff), "s"(s1) : "memory");
        asm volatile("global_load_async_to_lds_b32 %0, %1, %2"
                     :: "v"(l2), "v"(goff), "s"(s2) : "memory");
        asm volatile("global_load_async_to_lds_b32 %0, %1, %2"
                     :: "v"(l3), "v"(goff), "s"(s3) : "memory");
        asm volatile("s_wait_asynccnt 0" ::: "memory");
    }
    __syncthreads();

    const v4f* __restrict__ iv = (const v4f*)(img + (size_t)slice * kHW);
    v4f* __restrict__ o1 = (v4f*)(out + (size_t)slice * kHW);
    v4f* __restrict__ o2 = (v4f*)(out + (size_t)kTot + (size_t)slice * kHW);
    v4f* __restrict__ o3 = (v4f*)(out + 2 * (size_t)kTot + (size_t)slice * kHW);

    const float* L1 = lut;
    const float* L2 = lut + 256;
    const float* L3 = lut + 512;

    const int stride = gridDim.x * blockDim.x;          // 65,536 with grid.x=256
    for (int i = blockIdx.x * blockDim.x + t; i < kVecPerSlice; i += stride) {
        // Streaming read-once load: non-temporal, 128b per lane (coalesced 4KB/wave... 512B/wave32)
        v4f p = __builtin_nontemporal_load(&iv[i]);

        int i0 = __float2int_rn(p.x * 255.0f);
        int i1 = __float2int_rn(p.y * 255.0f);
        int i2 = __float2int_rn(p.z * 255.0f);
        int i3 = __float2int_rn(p.w * 255.0f);

        // 12 ds_load_b32 gathers from the 1KB tables
        v4f r1 = { L1[i0], L1[i1], L1[i2], L1[i3] };
        v4f r2 = { L2[i0], L2[i1], L2[i2], L2[i3] };
        v4f r3 = { L3[i0], L3[i1], L3[i2], L3[i3] };

        // Write-once streams: non-temporal B128 stores
        __builtin_nontemporal_store(r1, &o1[i]);
        __builtin_nontemporal_store(r2, &o2[i]);
        __builtin_nontemporal_store(r3, &o3[i]);
    }
}

extern "C" void kernel_launch(void* const* d_in, const int* in_sizes, int n_in,
                              void* d_out, int out_size, void* d_ws, size_t ws_size,
                              hipStream_t stream) {
    (void)in_sizes; (void)n_in; (void)out_size; (void)d_ws; (void)ws_size;
    const float* img = (const float*)d_in[0];
    const float* tf1 = (const float*)d_in[1];
    const float* tf2 = (const float*)d_in[2];
    const float* tf3 = (const float*)d_in[3];
    float* out = (float*)d_out;

    dim3 grid(256, kSlices, 1);   // 6144 blocks of 256 threads; 4 float4s per thread
    dim3 block(256, 1, 1);
    hipLaunchKernelGGL(IntensityTransformation_lut3_kernel, grid, block, 0, stream,
                       img, tf1, tf2, tf3, out);
}